// RegionLoss_9045201125616
// MI455X (gfx1250) — compile-verified
//
#include <hip/hip_runtime.h>
#include <hip/hip_bf16.h>
#include <math.h>

// ---------------- problem constants ----------------
#define NBATCH 16
#define KPTS   9
#define NCH    20          // 2*K + 1 + NC
#define NHW    5776        // 76*76
#define NGRID  76
#define MAXT   50
#define NLAB   21          // 2*K + 3
#define IMW    640.0f
#define IMH    480.0f
#define DIST_T 30.0f
#define SIL_T  0.6f
#define OBJ_S  5.0f
#define NOOBJ_S 0.1f
#define INV_EXPM1_2 0.15651764274966565f  // 1/(e^2 - 1)
#define NTILE  361         // 5776 / 16
#define MROWS  64          // 50 targets padded to 4 WMMA M-tiles

typedef __attribute__((ext_vector_type(2))) float v2f;
typedef __attribute__((ext_vector_type(8))) float v8f;

__device__ __forceinline__ float sigm(float x) {
  return 1.0f / (1.0f + __expf(-x));
}

// corner confidence from squared pixel distance
__device__ __forceinline__ float corner_c(float d2) {
  float dist = sqrtf(fmaxf(d2, 0.0f));
  float c = (__expf(2.0f * (1.0f - dist * (1.0f / DIST_T))) - 1.0f) * INV_EXPM1_2;
  return (dist < DIST_T) ? c : 0.0f;
}

// ---------------- kernel 1: target prep ----------------
// Builds WMMA A-matrix rows a = (gx*W, gy*H, |g|^2, 1) per (b, k, t); invalid/padded
// rows become (0,0,1e12,1) -> dist^2 huge -> conf 0. Also computes validity
// (cumprod of first-x != 0) and zeroes the loss accumulator.
__global__ void prep_kernel(const float* __restrict__ target,
                            float4* __restrict__ gA,
                            int* __restrict__ valid,
                            float* __restrict__ loss_out) {
  int b = blockIdx.x;
  int tid = threadIdx.x;                 // 64 threads
  __shared__ int svalid[MROWS];
  if (tid == 0) {
    int v = 1;
    for (int t = 0; t < MROWS; ++t) {
      if (t < MAXT) v = v && (target[b * (MAXT * NLAB) + t * NLAB + 1] != 0.0f);
      else v = 0;
      svalid[t] = v;
      valid[b * MROWS + t] = v;
    }
    if (b == 0) loss_out[0] = 0.0f;
  }
  __syncthreads();
  int t = tid;
  int v = svalid[t];
  const float* tg = target + b * (MAXT * NLAB) + t * NLAB + 1;
  for (int k = 0; k < KPTS; ++k) {
    float4 a;
    if (v) {
      float gx = tg[2 * k]     * IMW;
      float gy = tg[2 * k + 1] * IMH;
      a = make_float4(gx, gy, gx * gx + gy * gy, 1.0f);
    } else {
      a = make_float4(0.0f, 0.0f, 1.0e12f, 1.0f);
    }
    gA[(b * KPTS + k) * MROWS + t] = a;
  }
}

// ---------------- kernel 2: per-cell silence mask via WMMA ----------------
// One wave handles one 16-cell tile. For each of 4 M-tiles (64 padded targets)
// and 9 keypoints, V_WMMA_F32_16X16X4_F32 produces a 16x16 tile of squared
// pixel distances via the homogeneous 4-vector trick; VALU applies the
// exp-based confidence, sums over k, maxes over targets.
__global__ __launch_bounds__(256) void conf_mask_kernel(
    const float* __restrict__ out,
    const float4* __restrict__ gA,
    float* __restrict__ conf_mask,
    int* __restrict__ winner) {
  int b    = blockIdx.x;
  int wave = threadIdx.x >> 5;
  int lane = threadIdx.x & 31;
  int tile = blockIdx.y * 8 + wave;
  if (tile >= NTILE) return;             // wave-uniform: EXEC stays all-ones

  int m    = lane & 15;                  // N index within tile / A row within M-tile
  int half = lane >> 4;
  int c    = tile * 16 + m;              // cell id = gj*76 + gi
  int gi   = c % NGRID;
  int gj   = c / NGRID;

  const float* ob = out + (size_t)b * NCH * NHW + c;

  // Build B-matrix operands per keypoint:
  //   lanes 0-15  : rows K0 = -2*px*W, K1 = -2*py*H
  //   lanes 16-31 : rows K2 = 1,       K3 = |p|^2
  float bx[KPTS], by[KPTS], bsq[KPTS];
#pragma unroll
  for (int k = 0; k < KPTS; ++k) {
    float xv = ob[(2 * k)     * NHW];
    float yv = ob[(2 * k + 1) * NHW];
    if (k == 0) { xv = sigm(xv); yv = sigm(yv); }
    float pxW = (xv + (float)gi) * (IMW / (float)NGRID);
    float pyH = (yv + (float)gj) * (IMH / (float)NGRID);
    bx[k]  = -2.0f * pxW;
    by[k]  = -2.0f * pyH;
    bsq[k] = pxW * pxW + pyH * pyH;
  }

  float cellMax = 0.0f;
  for (int mt = 0; mt < 4; ++mt) {
    v8f acc = {};
#pragma unroll
    for (int k = 0; k < KPTS; ++k) {
      float4 ga = gA[(b * KPTS + k) * MROWS + mt * 16 + m];
      v2f a, bv;
      a.x  = half ? ga.z   : ga.x;       // A 16x4: lanes 0-15 K0,K1; 16-31 K2,K3
      a.y  = half ? ga.w   : ga.y;
      bv.x = half ? 1.0f   : bx[k];
      bv.y = half ? bsq[k] : by[k];
      v8f cz = {};
      v8f d = __builtin_amdgcn_wmma_f32_16x16x4_f32(
          /*neg_a=*/false, a, /*neg_b=*/false, bv,
          /*c_mod=*/(short)0, cz, /*reuse_a=*/false, /*reuse_b=*/false);
#pragma unroll
      for (int r = 0; r < 8; ++r) acc[r] += corner_c(d[r]);
    }
#pragma unroll
    for (int r = 0; r < 8; ++r) cellMax = fmaxf(cellMax, acc[r]);
  }
  // combine M rows 0-7 (lanes 0-15) with rows 8-15 (lanes 16-31) for same N
  cellMax = fmaxf(cellMax, __shfl_xor(cellMax, 16, 32));

  if (half == 0) {
    float cur = cellMax * (1.0f / (float)KPTS);  // mean over keypoints
    conf_mask[b * NHW + c] = (cur > SIL_T) ? 0.0f : NOOBJ_S;
    winner[b * NHW + c]    = -1;
  }
}

// ---------------- kernel 3: deterministic last-write-wins scatter ----------------
__global__ void scatter_kernel(const float* __restrict__ target,
                               const int* __restrict__ valid,
                               int* __restrict__ winner) {
  int b = threadIdx.x;
  if (b >= NBATCH) return;
  for (int t = 0; t < MAXT; ++t) {
    if (!valid[b * MROWS + t]) continue;
    float gx0 = target[b * (MAXT * NLAB) + t * NLAB + 1] * (float)NGRID;
    float gy0 = target[b * (MAXT * NLAB) + t * NLAB + 2] * (float)NGRID;
    int gi = (int)gx0;
    int gj = (int)gy0;
    if (gi >= 0 && gi < NGRID && gj >= 0 && gj < NGRID)
      winner[b * NHW + gj * NGRID + gi] = t;
  }
}

// ---------------- kernel 4: loss reduction ----------------
__global__ __launch_bounds__(256) void loss_kernel(
    const float* __restrict__ out,
    const float* __restrict__ target,
    const int* __restrict__ epoch_p,
    const float* __restrict__ conf_mask,
    const int* __restrict__ winner,
    float* __restrict__ loss_out) {
  int idx = blockIdx.x * 256 + threadIdx.x;
  float loss = 0.0f;
  if (idx < NBATCH * NHW) {
    int b = idx / NHW;
    int c = idx % NHW;
    const float* ob = out + (size_t)b * NCH * NHW + c;
    float conf = sigm(ob[(2 * KPTS) * NHW]);   // channel 18
    bool useConf = (*epoch_p > 15);
    int w = winner[idx];
    if (w < 0) {
      if (useConf) loss = 0.5f * conf_mask[idx] * conf * conf;  // tconf = 0
    } else {
      int gi = c % NGRID;
      int gj = c / NGRID;
      const float* tg = target + b * (MAXT * NLAB) + w * NLAB + 1;
      float lx = 0.0f, ly = 0.0f, ts = 0.0f;
#pragma unroll
      for (int k = 0; k < KPTS; ++k) {
        float xv = ob[(2 * k)     * NHW];
        float yv = ob[(2 * k + 1) * NHW];
        if (k == 0) { xv = sigm(xv); yv = sigm(yv); }
        float gx = tg[2 * k];
        float gy = tg[2 * k + 1];
        // coord targets relative to the winning cell (keypoint-0 cell == this cell)
        float tx = gx * (float)NGRID - (float)gi;
        float ty = gy * (float)NGRID - (float)gj;
        lx += (xv - tx) * (xv - tx);
        ly += (yv - ty) * (yv - ty);
        // tconf: distance in pixels between gt and this cell's prediction
        float dx = (gx - (xv + (float)gi) * (1.0f / (float)NGRID)) * IMW;
        float dy = (gy - (yv + (float)gj) * (1.0f / (float)NGRID)) * IMH;
        ts += corner_c(dx * dx + dy * dy);
      }
      loss = 0.5f * (lx + ly);                       // COORD_SCALE = 1
      if (useConf) {
        float d = conf - ts * (1.0f / (float)KPTS);
        loss += 0.5f * OBJ_S * d * d;                // conf_mask = 5 at winners
      }
    }
  }
  __shared__ float sd[256];
  sd[threadIdx.x] = loss;
  __syncthreads();
  for (int s = 128; s > 0; s >>= 1) {
    if (threadIdx.x < s) sd[threadIdx.x] += sd[threadIdx.x + s];
    __syncthreads();
  }
  if (threadIdx.x == 0) atomicAdd(loss_out, sd[0]);
}

// ---------------- host side ----------------
extern "C" void kernel_launch(void* const* d_in, const int* in_sizes, int n_in,
                              void* d_out, int out_size, void* d_ws, size_t ws_size,
                              hipStream_t stream) {
  const float* output = (const float*)d_in[0];
  const float* target = (const float*)d_in[1];
  const int*   epoch  = (const int*)d_in[2];
  float* loss_out     = (float*)d_out;

  // workspace layout (all 256B-aligned)
  char* ws = (char*)d_ws;
  size_t offA   = 0;                                             // float4 gA[NB][K][64]
  size_t szA    = (size_t)NBATCH * KPTS * MROWS * sizeof(float4);
  size_t offV   = (offA + szA + 255) & ~(size_t)255;             // int valid[NB][64]
  size_t szV    = (size_t)NBATCH * MROWS * sizeof(int);
  size_t offCM  = (offV + szV + 255) & ~(size_t)255;             // float conf_mask[NB][5776]
  size_t szCM   = (size_t)NBATCH * NHW * sizeof(float);
  size_t offWin = (offCM + szCM + 255) & ~(size_t)255;           // int winner[NB][5776]

  float4* gA       = (float4*)(ws + offA);
  int*    valid    = (int*)(ws + offV);
  float*  confmask = (float*)(ws + offCM);
  int*    winner   = (int*)(ws + offWin);

  // 1) target prep + zero loss accumulator
  prep_kernel<<<dim3(NBATCH), dim3(MROWS), 0, stream>>>(target, gA, valid, loss_out);

  // 2) silence mask via WMMA: 16 batches x 46 tile-groups, 8 waves/block
  conf_mask_kernel<<<dim3(NBATCH, (NTILE + 7) / 8), dim3(256), 0, stream>>>(
      output, gA, confmask, winner);

  // 3) deterministic scatter of winning targets
  scatter_kernel<<<dim3(1), dim3(32), 0, stream>>>(target, valid, winner);

  // 4) loss reduction
  loss_kernel<<<dim3((NBATCH * NHW) / 256), dim3(256), 0, stream>>>(
      output, target, epoch, confmask, winner, loss_out);
}